// Fuzzy_type2_4123168604216
// MI455X (gfx1250) — compile-verified
//
#include <hip/hip_runtime.h>

#define FP_EPS 1e-4f

// Fuzzy type-2 pooling of one 2x2 patch (x0,x1 = top row; x2,x3 = bottom row).
__device__ __forceinline__ float fuzzy_pool4(float x0, float x1, float x2, float x3) {
    // band means (P=2, N=4, HH=2): kmm0 = mean(x[1:3]), kmm1 = mean(all)
    float kmm0 = 0.5f  * (x1 + x2);
    float kmm1 = 0.25f * ((x0 + x1) + (x2 + x3));
    float vavg = 0.5f  * (kmm0 + kmm1);
    float o0 = fabsf(x0 - vavg);
    float o1 = fabsf(x1 - vavg);
    float o2 = fabsf(x2 - vavg);
    float o3 = fabsf(x3 - vavg);
    float s0 = 0.5f  * (o1 + o2) + FP_EPS;
    float s1 = 0.25f * ((o0 + o1) + (o2 + o3)) + FP_EPS;
    float i0 = 1.0f / s0;
    float i1 = 1.0f / s1;
    const float C = -0.72134752044448170368f; // -0.5 * log2(e): exp(-0.5 d^2) = 2^(C d^2)
    float d;
    d = (x0 - kmm0) * i0; float p00 = __builtin_amdgcn_exp2f(C * d * d);
    d = (x1 - kmm0) * i0; float p01 = __builtin_amdgcn_exp2f(C * d * d);
    d = (x2 - kmm0) * i0; float p02 = __builtin_amdgcn_exp2f(C * d * d);
    d = (x3 - kmm0) * i0; float p03 = __builtin_amdgcn_exp2f(C * d * d);
    d = (x0 - kmm1) * i1; float p10 = __builtin_amdgcn_exp2f(C * d * d);
    d = (x1 - kmm1) * i1; float p11 = __builtin_amdgcn_exp2f(C * d * d);
    d = (x2 - kmm1) * i1; float p12 = __builtin_amdgcn_exp2f(C * d * d);
    d = (x3 - kmm1) * i1; float p13 = __builtin_amdgcn_exp2f(C * d * d);
    // thresh = min_n max_m pi[m][n]
    float thresh = fminf(fminf(fmaxf(p00, p10), fmaxf(p01, p11)),
                         fminf(fmaxf(p02, p12), fmaxf(p03, p13)));
    float a0 = 0.5f * (p00 + p10);
    float a1 = 0.5f * (p01 + p11);   // avg_pi[..., HH-1]
    float a2 = 0.5f * (p02 + p12);
    float a3 = 0.5f * (p03 + p13);
    float num = a0 * x0 + a1 * x1 + a2 * x2 + a3 * x3;
    float den = (a0 + a1) + (a2 + a3);
    float denoised = num / den;
    float r = (s1 < 0.001f) ? vavg : denoised;   // secondary -> v_avg, else denoised
    return (a1 >= thresh) ? kmm1 : r;            // primary  -> mean_x (== kmm1)
}

// Stream: input = 8 KiB chunks (4 contiguous row-pairs), double-buffered async
// copies into LDS, compute 2 output pixels per thread, coalesced float2 stores.
__global__ __launch_bounds__(256)
void fuzzy_pool_kernel(const float* __restrict__ x, float* __restrict__ out, int nchunks) {
    __shared__ __align__(16) float smem[2][2048];   // 2 x 8 KiB double buffer
    const int tid = threadIdx.x;

    // Per-lane LDS byte addresses for the async DMA destination (low 32 bits of
    // a generic LDS pointer are the wave-relative LDS byte address).
    unsigned lds0 = (unsigned)(size_t)(&smem[0][0]) + (unsigned)tid * 16u;
    unsigned lds1 = (unsigned)(size_t)(&smem[1][0]) + (unsigned)tid * 16u;

    int c = (int)blockIdx.x;
    int buf = 0;
    if (c < nchunks) {
        const char* g = (const char*)x + ((size_t)c << 13) + (size_t)tid * 16;
        asm volatile("global_load_async_to_lds_b128 %0, %1, off\n\t"
                     "global_load_async_to_lds_b128 %0, %1, off offset:4096"
                     :: "v"(lds0), "v"(g) : "memory");
    }
    for (; c < nchunks; c += (int)gridDim.x) {
        int nc = c + (int)gridDim.x;
        if (nc < nchunks) {
            // prefetch next chunk into the other buffer, then wait for current
            const char* g = (const char*)x + ((size_t)nc << 13) + (size_t)tid * 16;
            unsigned dst = (buf ? lds0 : lds1);
            asm volatile("global_load_async_to_lds_b128 %0, %1, off\n\t"
                         "global_load_async_to_lds_b128 %0, %1, off offset:4096"
                         :: "v"(dst), "v"(g) : "memory");
            asm volatile("s_wait_asynccnt 0x2" ::: "memory");
        } else {
            asm volatile("s_wait_asynccnt 0x0" ::: "memory");
        }
        __syncthreads();   // all waves' portions of chunk c are now in LDS buffer smem[buf]

        const int rp = tid >> 6;   // row-pair within chunk (0..3)
        const int u  = tid & 63;   // float4 unit within the row (0..63)
        const float* base = &smem[buf][rp * 512 + u * 4];
        float4 a = *(const float4*)(base);        // top input row, cols 4u..4u+3
        float4 b = *(const float4*)(base + 256);  // bottom input row
        float2 r;
        r.x = fuzzy_pool4(a.x, a.y, b.x, b.y);
        r.y = fuzzy_pool4(a.z, a.w, b.z, b.w);
        *(float2*)(out + ((size_t)c << 9) + (size_t)tid * 2) = r;

        buf ^= 1;
        __syncthreads();   // protect LDS buffer smem[buf^1] before next iteration overwrites it
    }
}

extern "C" void kernel_launch(void* const* d_in, const int* in_sizes, int n_in,
                              void* d_out, int out_size, void* d_ws, size_t ws_size,
                              hipStream_t stream) {
    (void)n_in; (void)out_size; (void)d_ws; (void)ws_size;
    const float* x = (const float*)d_in[0];
    float* out = (float*)d_out;
    long long n = (long long)in_sizes[0];        // 8*64*256*256 = 33,554,432 floats
    int nchunks = (int)(n / 2048);               // 8 KiB input chunks -> 16384
    int grid = nchunks < 2048 ? nchunks : 2048;  // ~8 chunks per block, pipelined
    if (grid < 1) grid = 1;
    fuzzy_pool_kernel<<<grid, 256, 0, stream>>>(x, out, nchunks);
}